// LinearMimo_9208409882670
// MI455X (gfx1250) — compile-verified
//
#include <hip/hip_runtime.h>

// MIMO IIR filter bank: y[b,t,o] = sum_i IIR3(FIR3(u[b, t-1, i]))  (NK=1 delay)
// Layout: lane = input channel i, wave = output channel o (8 per block),
// block = (batch b, o-group). 128 blocks x 256 threads (8 wave32 waves).

typedef float v2f __attribute__((ext_vector_type(2)));
typedef float v8f __attribute__((ext_vector_type(8)));

constexpr int kT    = 2048;
constexpr int kCIN  = 32;
constexpr int kCOUT = 32;
constexpr int kTT   = 16;   // timesteps per tile (= WMMA M)
constexpr int kOPB  = 8;    // output channels (waves) per block
constexpr int kPAD  = 17;   // LDS row pitch (bank-conflict-free)

__global__ __launch_bounds__(256) void mimo_iir_wmma_kernel(
    const float* __restrict__ u,    // (B, T, CIN)
    const float* __restrict__ bc,   // (COUT, CIN, 3)
    const float* __restrict__ ac,   // (COUT, CIN, 3)
    float* __restrict__ out)        // (B, T, COUT)
{
    __shared__ float u_lds[kTT * kCIN];          // 2 KB
    __shared__ float y_lds[kOPB * kCIN * kPAD];  // ~17 KB, per-wave private slabs

    const int tid  = threadIdx.x;
    const int w    = tid >> 5;   // local wave -> local output channel
    const int lane = tid & 31;   // input channel i
    const int b    = blockIdx.x / (kCOUT / kOPB);
    const int o    = (blockIdx.x % (kCOUT / kOPB)) * kOPB + w;
    const int i    = lane;

    // Per-thread filter coefficients
    const int cb = (o * kCIN + i) * 3;
    const float b0 = bc[cb + 0], b1 = bc[cb + 1], b2 = bc[cb + 2];
    const float a0 = ac[cb + 0], a1 = ac[cb + 1], a2 = ac[cb + 2];

    // Rolling state: up1=u[t-1], up2=u[t-2], up3=u[t-3]; y1..y3 = y[t-1..t-3]
    float up1 = 0.f, up2 = 0.f, up3 = 0.f;
    float y1 = 0.f, y2 = 0.f, y3 = 0.f;

    const float* ub = u   + (size_t)b * kT * kCIN;
    float*       ob = out + (size_t)b * kT * kCOUT;

    // A-fragment addressing for v_wmma_f32_16x16x4_f32 (16x4, M=lane&15):
    //   VGPR0: lanes 0-15 -> K=0, lanes 16-31 -> K=2 ; VGPR1: K=1 / K=3
    const int m   = lane & 15;
    const int khi = (lane < 16) ? 0 : 2;

    v2f bfrag; bfrag.x = 1.0f; bfrag.y = 1.0f;   // all-ones 4x16 B matrix

    float* yw = &y_lds[w * kCIN * kPAD];         // this wave's private slab

    for (int t0 = 0; t0 < kT; t0 += kTT) {
        __syncthreads();  // protect u_lds from previous tile's readers
        // Coalesced stage of u[b, t0:t0+16, :] (512 contiguous floats)
        u_lds[tid]       = ub[t0 * kCIN + tid];
        u_lds[tid + 256] = ub[t0 * kCIN + tid + 256];
        __syncthreads();

        // 16 recurrence steps; each lane writes its y_t column into LDS
        #pragma unroll
        for (int tt = 0; tt < kTT; ++tt) {
            // FIR: x = b0*u_d[t] + b1*u_d[t-1] + b2*u_d[t-2], u_d[t] = u[t-1]
            float x  = fmaf(b0, up1, fmaf(b1, up2, b2 * up3));
            // IIR: terms on older state first -> 1-FMA critical path on y1
            float p  = fmaf(-a1, y2, fmaf(-a2, y3, x));
            float yt = fmaf(-a0, y1, p);
            yw[i * kPAD + tt] = yt;
            y3 = y2; y2 = y1; y1 = yt;
            up3 = up2; up2 = up1; up1 = u_lds[tt * kCIN + i];
        }
        __builtin_amdgcn_wave_barrier();  // same-wave LDS RAW ordering hint

        // Reduce over the 32 input channels on the matrix pipe:
        // D[m,n] = sum_g sum_k Y[t=m, i=4g+k] * 1  (chained through C)
        v8f acc = {};
        #pragma unroll
        for (int g = 0; g < 8; ++g) {
            v2f afrag;
            afrag.x = yw[(4 * g + khi + 0) * kPAD + m];
            afrag.y = yw[(4 * g + khi + 1) * kPAD + m];
            acc = __builtin_amdgcn_wmma_f32_16x16x4_f32(
                false, afrag, false, bfrag, (short)0, acc, false, false);
        }

        // D layout: VGPR r -> M=r (lanes 0-15) / M=8+r (lanes 16-31), N replicated
        if (lane == 0) {
            #pragma unroll
            for (int r = 0; r < 8; ++r) ob[(t0 + r) * kCOUT + o] = acc[r];
        } else if (lane == 16) {
            #pragma unroll
            for (int r = 0; r < 8; ++r) ob[(t0 + 8 + r) * kCOUT + o] = acc[r];
        }
    }
}

extern "C" void kernel_launch(void* const* d_in, const int* in_sizes, int n_in,
                              void* d_out, int out_size, void* d_ws, size_t ws_size,
                              hipStream_t stream) {
    (void)n_in; (void)out_size; (void)d_ws; (void)ws_size;
    const float* u  = (const float*)d_in[0];
    const float* bc = (const float*)d_in[1];
    const float* ac = (const float*)d_in[2];
    float* out = (float*)d_out;

    const int B = in_sizes[0] / (kT * kCIN);   // = 32
    dim3 grid(B * (kCOUT / kOPB));             // 128 blocks
    dim3 block(kOPB * 32);                     // 256 threads = 8 wave32 waves
    mimo_iir_wmma_kernel<<<grid, block, 0, stream>>>(u, bc, ac, out);
}